// MultiheadCrossAttention_584115552473
// MI455X (gfx1250) — compile-verified
//
#include <hip/hip_runtime.h>
#include <math.h>

// Problem constants (match reference)
#define BB   64
#define TT   1024
#define DD   512    // ENC_D == DEC_D == PROJ_D
#define HH   4
#define DHH  128    // DD / HH
#define SCALE 11.313708498984761f   // sqrt(DH) — reference multiplies scores by sqrt(dh)

typedef float v2f __attribute__((ext_vector_type(2)));
typedef float v4f __attribute__((ext_vector_type(4)));
typedef float v8f __attribute__((ext_vector_type(8)));

// ---------------------------------------------------------------------------
// Kernel 1: per (b,h): q_h = Wq_h dec_h[b] + bq_h  (kept in LDS), then
//   qk[b,h,e]  = SCALE * sum_j q_h[j] * Wk[h*DH+j, e]
//   sbias[b,h] = SCALE * q_h · bk_h
// ---------------------------------------------------------------------------
__global__ __launch_bounds__(DHH) void qk_kernel(
    const float* __restrict__ dec_h, const float* __restrict__ Wq,
    const float* __restrict__ bq,    const float* __restrict__ Wk,
    const float* __restrict__ bk,    float* __restrict__ qk,
    float* __restrict__ sbias) {
  const int b = blockIdx.x, h = blockIdx.y, j = threadIdx.x;  // j in [0,128)
  __shared__ float qs[DHH];
  __shared__ float red[DHH];

  const int row = h * DHH + j;
  const float* w = Wq + (size_t)row * DD;
  const float* x = dec_h + (size_t)b * DD;
  float acc = bq[row];
  for (int e = 0; e < DD; ++e) acc += w[e] * x[e];
  qs[j]  = acc;
  red[j] = acc * bk[row];
  __syncthreads();
  for (int s = DHH / 2; s > 0; s >>= 1) {
    if (j < s) red[j] += red[j + s];
    __syncthreads();
  }
  if (j == 0) sbias[b * HH + h] = red[0] * SCALE;

  // each thread computes 4 columns of qk (coalesced over e)
  for (int c = 0; c < 4; ++c) {
    const int e = j + c * DHH;
    float s = 0.f;
    for (int jj = 0; jj < DHH; ++jj)
      s += qs[jj] * Wk[(size_t)(h * DHH + jj) * DD + e];
    qk[((size_t)b * HH + h) * DD + e] = s * SCALE;
  }
}

// ---------------------------------------------------------------------------
// Kernel 2: one workgroup per batch.
//   Pass A: s[h][t] = enc_t · qk_h + sbias  (float4 coalesced + shuffle reduce)
//   Softmax with length mask (exact vs reference: exp(FLT_MIN - m) == 0)
//   Pass B: ctx[h,e] = sum_t p[h][t] * enc[t][e] via V_WMMA_F32_16X16X4_F32,
//           M = heads padded to 16 (zero row in LDS), N = 16 e-cols, K = T.
// ---------------------------------------------------------------------------
__global__ __launch_bounds__(512) void attn_ctx_kernel(
    const float* __restrict__ enc, const int* __restrict__ enc_l,
    const float* __restrict__ qk,  const float* __restrict__ sbias,
    float* __restrict__ ctx) {
  const int b = blockIdx.x;
  const int tid = threadIdx.x;
  const int lane = tid & 31, wave = tid >> 5;  // 16 waves of 32

  __shared__ __align__(16) float qk_s[HH * DD];       // 8 KB
  __shared__ __align__(16) float scp[(HH + 1) * TT];  // 20 KB, row HH stays 0
  __shared__ float red[512];                          // 2 KB

  for (int i = tid; i < HH * DD; i += 512)
    qk_s[i] = qk[(size_t)b * HH * DD + i];
  for (int i = tid; i < TT; i += 512)
    scp[HH * TT + i] = 0.0f;                          // zero pad row for WMMA
  __syncthreads();

  const float* encb = enc + (size_t)b * TT * DD;
  const int len = enc_l[b];

  // ---- Pass A: each wave owns 64 consecutive t rows; b128 loads ----
  for (int t = wave * 64; t < wave * 64 + 64; ++t) {
    const float* row = encb + (size_t)t * DD;
    if (t + 1 < TT) __builtin_prefetch(row + DD, 0, 0);
    float a0 = 0.f, a1 = 0.f, a2 = 0.f, a3 = 0.f;
    for (int i = 0; i < DD / 128; ++i) {       // 4 iterations, 128 floats each
      const int e = i * 128 + lane * 4;
      const v4f x  = *(const v4f*)(row  + e);  // global_load_b128, coalesced
      const v4f k0 = *(const v4f*)(qk_s + 0 * DD + e);
      const v4f k1 = *(const v4f*)(qk_s + 1 * DD + e);
      const v4f k2 = *(const v4f*)(qk_s + 2 * DD + e);
      const v4f k3 = *(const v4f*)(qk_s + 3 * DD + e);
      a0 += x.x * k0.x + x.y * k0.y + x.z * k0.z + x.w * k0.w;
      a1 += x.x * k1.x + x.y * k1.y + x.z * k1.z + x.w * k1.w;
      a2 += x.x * k2.x + x.y * k2.y + x.z * k2.z + x.w * k2.w;
      a3 += x.x * k3.x + x.y * k3.y + x.z * k3.z + x.w * k3.w;
    }
    for (int off = 16; off > 0; off >>= 1) {
      a0 += __shfl_xor(a0, off, 32);
      a1 += __shfl_xor(a1, off, 32);
      a2 += __shfl_xor(a2, off, 32);
      a3 += __shfl_xor(a3, off, 32);
    }
    if (lane == 0) {
      scp[0 * TT + t] = a0; scp[1 * TT + t] = a1;
      scp[2 * TT + t] = a2; scp[3 * TT + t] = a3;
    }
  }
  __syncthreads();

  // ---- Softmax per head (mask t >= len) ----
  for (int h = 0; h < HH; ++h) {
    const float sb = sbias[(size_t)b * HH + h];
    float m = -3.4e38f;
    for (int t = tid; t < len; t += 512) m = fmaxf(m, scp[h * TT + t] + sb);
    red[tid] = m; __syncthreads();
    for (int s = 256; s > 0; s >>= 1) {
      if (tid < s) red[tid] = fmaxf(red[tid], red[tid + s]);
      __syncthreads();
    }
    m = red[0]; __syncthreads();
    float l = 0.f;
    for (int t = tid; t < len; t += 512) {
      const float p = __expf(scp[h * TT + t] + sb - m);
      scp[h * TT + t] = p;
      l += p;
    }
    red[tid] = l; __syncthreads();
    for (int s = 256; s > 0; s >>= 1) {
      if (tid < s) red[tid] += red[tid + s];
      __syncthreads();
    }
    const float inv = 1.0f / red[0]; __syncthreads();
    for (int t = tid; t < TT; t += 512)
      scp[h * TT + t] = (t < len) ? scp[h * TT + t] * inv : 0.0f;
    __syncthreads();
  }

  // ---- Pass B: ctx via fp32 WMMA, 2-stage software pipeline ----
  const int ml   = lane & 15;            // A: M row (head, padded); B: N col
  const int ksel = (lane >> 4) << 1;     // lanes 0-15 -> K 0,1 ; 16-31 -> K 2,3
  const int mrow = (ml < HH) ? ml : HH;  // padded heads read the LDS zero row
  const float* aptr = scp + (size_t)mrow * TT;  // address select, no EXEC mask

  for (int et = wave; et < DD / 16; et += 16) {
    const int e0 = et * 16;
    const float* base = encb + (size_t)ksel * DD + e0 + ml;  // B col for lane
    v8f c = {};
    // prologue: fragments for kt = 0
    v2f a0 = *(const v2f*)(aptr + ksel);               // ds_load_b64
    v2f b0; b0.x = base[0]; b0.y = base[DD];
    for (int kt = 0; kt < TT - 4; kt += 4) {
      // prefetch next iteration's fragments before issuing this WMMA
      v2f a1 = *(const v2f*)(aptr + kt + 4 + ksel);
      v2f b1;
      b1.x = base[(size_t)(kt + 4) * DD];
      b1.y = base[(size_t)(kt + 5) * DD];
      c = __builtin_amdgcn_wmma_f32_16x16x4_f32(false, a0, false, b0,
                                                (short)0, c, false, false);
      a0 = a1; b0 = b1;
    }
    c = __builtin_amdgcn_wmma_f32_16x16x4_f32(false, a0, false, b0,
                                              (short)0, c, false, false);
    // C/D layout: VGPR i -> M=i (lanes 0-15), M=i+8 (lanes 16-31); N = lane&15
    if (lane < 16) {
      for (int h = 0; h < HH; ++h)
        ctx[((size_t)b * HH + h) * DD + e0 + lane] = c[h];
    }
  }
}

// ---------------------------------------------------------------------------
// Kernel 3: att[b, h*DH+d] = bv + ctx[b,h,:] · Wv[h*DH+d, :]
// ---------------------------------------------------------------------------
__global__ __launch_bounds__(DHH) void att_kernel(
    const float* __restrict__ ctx, const float* __restrict__ Wv,
    const float* __restrict__ bv,  float* __restrict__ att) {
  const int bh = blockIdx.x;
  const int b = bh / HH, h = bh % HH;
  __shared__ float cs[DD];
  for (int i = threadIdx.x; i < DD; i += DHH) cs[i] = ctx[(size_t)bh * DD + i];
  __syncthreads();
  const int row = h * DHH + threadIdx.x;
  const float* w = Wv + (size_t)row * DD;
  float acc = bv[row];
  for (int e = 0; e < DD; ++e) acc += cs[e] * w[e];
  att[(size_t)b * DD + row] = acc;
}

// ---------------------------------------------------------------------------
// Kernel 4: out = att(64x512) @ Wf^T + bf, fp32 WMMA GEMM.
// grid (4 M-tiles, 4) x 256 threads (8 waves, each one 16x16 N-tile).
// ---------------------------------------------------------------------------
__global__ __launch_bounds__(256) void out_kernel(
    const float* __restrict__ att, const float* __restrict__ Wf,
    const float* __restrict__ bfv, float* __restrict__ out) {
  const int lane = threadIdx.x & 31, wave = threadIdx.x >> 5;
  const int m0 = blockIdx.x * 16;
  const int n0 = (blockIdx.y * 8 + wave) * 16;
  const int ml   = lane & 15;
  const int ksel = (lane >> 4) << 1;

  const float* arow = att + (size_t)(m0 + ml) * DD + ksel;  // A: row per lane
  const float* brow = Wf  + (size_t)(n0 + ml) * DD + ksel;  // B[k][n] = Wf[n][k]

  v8f c = {};
  v2f a0 = *(const v2f*)(arow);
  v2f b0 = *(const v2f*)(brow);
  for (int kt = 0; kt < DD - 4; kt += 4) {
    v2f a1 = *(const v2f*)(arow + kt + 4);
    v2f b1 = *(const v2f*)(brow + kt + 4);
    c = __builtin_amdgcn_wmma_f32_16x16x4_f32(false, a0, false, b0,
                                              (short)0, c, false, false);
    a0 = a1; b0 = b1;
  }
  c = __builtin_amdgcn_wmma_f32_16x16x4_f32(false, a0, false, b0,
                                            (short)0, c, false, false);

  const int mbase = (lane >> 4) * 8;
  const int n = n0 + ml;
  const float bias = bfv[n];
  for (int i = 0; i < 8; ++i)
    out[(size_t)(m0 + mbase + i) * DD + n] = c[i] + bias;
}

// ---------------------------------------------------------------------------
extern "C" void kernel_launch(void* const* d_in, const int* in_sizes, int n_in,
                              void* d_out, int out_size, void* d_ws, size_t ws_size,
                              hipStream_t stream) {
  const float* enc_h = (const float*)d_in[0];
  const int*   enc_l = (const int*)  d_in[1];
  const float* dec_h = (const float*)d_in[2];
  const float* Wk    = (const float*)d_in[3];
  const float* bk    = (const float*)d_in[4];
  const float* Wv    = (const float*)d_in[5];
  const float* bv    = (const float*)d_in[6];
  const float* Wq    = (const float*)d_in[7];
  const float* bq    = (const float*)d_in[8];
  const float* Wf    = (const float*)d_in[9];
  const float* bf    = (const float*)d_in[10];
  float* out = (float*)d_out;

  // workspace layout (floats): qk | sbias | ctx | att  (~1.2 MB total)
  float* qk    = (float*)d_ws;
  float* sbias = qk    + (size_t)BB * HH * DD;
  float* ctx   = sbias + (size_t)BB * HH;
  float* att   = ctx   + (size_t)BB * HH * DD;

  qk_kernel<<<dim3(BB, HH), DHH, 0, stream>>>(dec_h, Wq, bq, Wk, bk, qk, sbias);
  attn_ctx_kernel<<<BB, 512, 0, stream>>>(enc_h, enc_l, qk, sbias, ctx);
  att_kernel<<<BB * HH, DHH, 0, stream>>>(ctx, Wv, bv, att);
  out_kernel<<<dim3(4, 4), 256, 0, stream>>>(att, Wf, bf, out);
}